// SelfAttention_8710193676523
// MI455X (gfx1250) — compile-verified
//
#include <hip/hip_runtime.h>
#include <hip/hip_bf16.h>

// B=32, TL=4096, TG=1, K=128, H=8
#define BATCH 32
#define TL    4096
#define KDIM  128
#define HEADS 8

typedef __attribute__((ext_vector_type(2))) float v2f;
typedef __attribute__((ext_vector_type(8))) float v8f;

// ---------------------------------------------------------------------------
// Kernel A: q_raw[b,h,c] = x_glob[b,:] @ Wq[:, h*128+c]
//           wtil[b,h,j]  = K^-0.5 * sum_c Wk[j, h*128+c] * q_raw[b,h,c]
// grid = 32 (one block per b), 256 threads
// ---------------------------------------------------------------------------
__global__ void qw_kernel(const float* __restrict__ xg, const float* __restrict__ Wq,
                          const float* __restrict__ Wk, float* __restrict__ wtil) {
    int b = blockIdx.x;
    int t = threadIdx.x;
    __shared__ float xs[KDIM];
    __shared__ float qs[HEADS * KDIM];
    if (t < KDIM) xs[t] = xg[b * KDIM + t];
    __syncthreads();
    for (int r = 0; r < 4; ++r) {
        int hc = t + r * 256;                       // 0..1023
        float acc = 0.f;
        for (int j = 0; j < KDIM; ++j)
            acc = fmaf(xs[j], Wq[j * (HEADS * KDIM) + hc], acc);
        qs[hc] = acc;
    }
    __syncthreads();
    const float sc = 0.08838834764831845f;          // 128^-0.5  (= scale^2)
    for (int r = 0; r < 4; ++r) {
        int f = t + r * 256;                        // h*128 + j
        int h = f >> 7, j = f & 127;
        const float* wkrow = Wk + j * (HEADS * KDIM) + h * KDIM;
        const float* qrow  = qs + h * KDIM;
        float acc = 0.f;
        for (int c = 0; c < KDIM; ++c) acc = fmaf(wkrow[c], qrow[c], acc);
        wtil[b * (HEADS * KDIM) + f] = acc * sc;
    }
}

// ---------------------------------------------------------------------------
// Kernel B (hot pass 1): logits[b,h,l] = x_local[b,l,:] . wtil[b,h,:]
// Skinny GEMM per b: (4096 x 128) @ (128 x 8->pad16) via V_WMMA_F32_16X16X4_F32.
// grid = (32 l-chunks, 32 b), 256 threads = 8 waves, wave -> 16-row tile.
// ---------------------------------------------------------------------------
__global__ void logits_kernel(const float* __restrict__ x, const float* __restrict__ wtil,
                              float* __restrict__ logits) {
    int b     = blockIdx.y;
    int chunk = blockIdx.x;
    int t     = threadIdx.x;
    int wave  = t >> 5, lane = t & 31;

    // Stage wtil[b] in LDS, zero-padded to 16 "heads" so the B operand needs no branch.
    __shared__ float wl[16 * KDIM];
    for (int r = 0; r < 8; ++r) {
        int idx = t + r * 256;
        wl[idx] = (idx < HEADS * KDIM) ? wtil[b * (HEADS * KDIM) + idx] : 0.f;
    }
    __syncthreads();

    int l0   = chunk * 128 + wave * 16;
    int m    = lane & 15;                 // A row (lanes 0-15 / 16-31 both map M=0..15)
    int koff = (lane < 16) ? 0 : 2;       // ISA A-16x4 layout: halves hold K pairs {0,1}/{2,3}
    int n    = m;                         // B/D column = head index (valid < 8)

    const float* xrow = x + ((size_t)b * TL + l0 + m) * KDIM;
    __builtin_prefetch(xrow, 0, 3);

    v8f c = {};
    for (int kk = 0; kk < 32; ++kk) {
        int k0 = kk * 4 + koff;
        v2f a;                             // A[M=m, K=k0], A[M=m, K=k0+1]
        a.x = xrow[k0];
        a.y = xrow[k0 + 1];
        v2f bf;                            // B[K=k0, N=n] = wtil[n, k0]
        bf.x = wl[n * KDIM + k0];
        bf.y = wl[n * KDIM + k0 + 1];
        c = __builtin_amdgcn_wmma_f32_16x16x4_f32(false, a, false, bf,
                                                  (short)0, c, false, false);
    }
    // C layout: VGPR i -> rows (l0+i | l0+8+i), N = lane (h)
    if (n < HEADS) {
        int base = l0 + ((lane < 16) ? 0 : 8);
        float* lp = logits + ((size_t)(b * HEADS + n)) * TL + base;
#pragma unroll
        for (int i = 0; i < 8; ++i) lp[i] = c[i];
    }
}

// ---------------------------------------------------------------------------
// Kernel C: per-(b,h) softmax stats over 4096 logits -> (max, sum exp)
// grid = 256 blocks (b*8+h), 256 threads
// ---------------------------------------------------------------------------
__global__ void softmax_stats_kernel(const float* __restrict__ logits, float2* __restrict__ msum) {
    int bh = blockIdx.x;
    int t  = threadIdx.x;
    const float* lp = logits + (size_t)bh * TL;
    __shared__ float red[256];

    float mx = -__builtin_inff();
    for (int i = t; i < TL; i += 256) mx = fmaxf(mx, lp[i]);
    red[t] = mx;
    __syncthreads();
    for (int s = 128; s > 0; s >>= 1) {
        if (t < s) red[t] = fmaxf(red[t], red[t + s]);
        __syncthreads();
    }
    mx = red[0];
    __syncthreads();

    float sum = 0.f;
    for (int i = t; i < TL; i += 256) sum += __expf(lp[i] - mx);
    red[t] = sum;
    __syncthreads();
    for (int s = 128; s > 0; s >>= 1) {
        if (t < s) red[t] += red[t + s];
        __syncthreads();
    }
    if (t == 0) msum[bh] = make_float2(mx, red[0]);
}

// ---------------------------------------------------------------------------
// Kernel D (hot pass 2): s[b,h,:] = sum_l attn[b,h,l] * x_local[b,l,:]
// GEMM per b: (8->pad16 x 4096) @ (4096 x 128) via V_WMMA_F32_16X16X4_F32,
// K split 8 ways across blocks; partials to disjoint slices (deterministic).
// grid = (8 k-splits, 32 b), 256 threads = 8 waves, wave -> 16-col tile.
// x_local is re-read here from L2 (192MB L2 >> 64MB tensor).
// ---------------------------------------------------------------------------
__global__ void svec_kernel(const float* __restrict__ x, const float* __restrict__ logits,
                            const float2* __restrict__ msum, float* __restrict__ svec) {
    int b  = blockIdx.y;
    int ls = blockIdx.x;                  // l-split: 512 rows each
    int t  = threadIdx.x;
    int wave = t >> 5, lane = t & 31;

    int j0   = wave * 16;                 // output feature tile
    int m    = lane & 15;                 // A row = head (valid < 8)
    int koff = (lane < 16) ? 0 : 2;
    bool hv  = m < HEADS;

    float hmax = hv ? msum[b * HEADS + m].x : 0.f;
    const float* lrow = logits + ((size_t)(b * HEADS + (hv ? m : 0))) * TL;
    const float* xb   = x + (size_t)b * TL * KDIM;
    int lbase = ls * 512;

    v8f c = {};
    for (int kk = 0; kk < 128; ++kk) {
        int l = lbase + kk * 4 + koff;
        v2f a;                             // A[M=h, K=l] = exp(logit - max), pad rows = 0
        a.x = hv ? __expf(lrow[l]     - hmax) : 0.f;
        a.y = hv ? __expf(lrow[l + 1] - hmax) : 0.f;
        v2f bf;                            // B[K=l, N=n] = x[b, l, j0+n]
        bf.x = xb[(size_t)l       * KDIM + j0 + m];
        bf.y = xb[(size_t)(l + 1) * KDIM + j0 + m];
        c = __builtin_amdgcn_wmma_f32_16x16x4_f32(false, a, false, bf,
                                                  (short)0, c, false, false);
    }
    // C layout: lanes 0-15 hold rows M=i (= head i, all valid), N = lane -> j = j0+lane
    if (lane < 16) {
#pragma unroll
        for (int i = 0; i < HEADS; ++i) {
            float inv = 1.f / msum[b * HEADS + i].y;
            svec[(((size_t)ls * BATCH + b) * HEADS + i) * KDIM + j0 + lane] = c[i] * inv;
        }
    }
}

// ---------------------------------------------------------------------------
// Kernel E: reduce 8 partials; out[hc] = s[b,h,:] @ Wv[:, hc]; y = out @ Wu + bu
// grid = 32 (per b), 256 threads
// ---------------------------------------------------------------------------
__global__ void out_kernel(const float* __restrict__ svec, const float* __restrict__ Wv,
                           const float* __restrict__ Wu, const float* __restrict__ bu,
                           float* __restrict__ y) {
    int b = blockIdx.x;
    int t = threadIdx.x;
    __shared__ float sl[HEADS * KDIM];
    __shared__ float ol[HEADS * KDIM];
    for (int r = 0; r < 4; ++r) {
        int f = t + r * 256;
        float acc = 0.f;
#pragma unroll
        for (int p = 0; p < 8; ++p)
            acc += svec[((size_t)p * BATCH + b) * (HEADS * KDIM) + f];
        sl[f] = acc;
    }
    __syncthreads();
    for (int r = 0; r < 4; ++r) {
        int f = t + r * 256;
        int h = f >> 7, cidx = f & 127;
        const float* srow = sl + h * KDIM;
        float acc = 0.f;
        for (int j = 0; j < KDIM; ++j)
            acc = fmaf(srow[j], Wv[j * (HEADS * KDIM) + h * KDIM + cidx], acc);
        ol[f] = acc;
    }
    __syncthreads();
    if (t < KDIM) {
        float acc = bu[t];
        for (int q = 0; q < HEADS * KDIM; ++q)
            acc = fmaf(ol[q], Wu[q * KDIM + t], acc);
        y[b * KDIM + t] = acc;
    }
}

// ---------------------------------------------------------------------------
extern "C" void kernel_launch(void* const* d_in, const int* in_sizes, int n_in,
                              void* d_out, int out_size, void* d_ws, size_t ws_size,
                              hipStream_t stream) {
    const float* x_local = (const float*)d_in[0];
    const float* x_glob  = (const float*)d_in[1];
    const float* Wq      = (const float*)d_in[2];
    const float* Wk      = (const float*)d_in[3];
    const float* Wv      = (const float*)d_in[4];
    const float* Wu      = (const float*)d_in[5];
    const float* bu      = (const float*)d_in[6];
    float* y = (float*)d_out;

    // Workspace layout (floats): wtil 32K | logits 1M | msum 512 | svec partials 256K
    float*  ws     = (float*)d_ws;
    float*  wtil   = ws;                                   // 32*8*128      = 32768
    float*  logits = wtil + BATCH * HEADS * KDIM;          // 32*8*4096     = 1048576
    float2* msum   = (float2*)(logits + (size_t)BATCH * HEADS * TL);   // 256 float2
    float*  svec   = (float*)(msum + BATCH * HEADS);       // 8*32*8*128    = 262144

    qw_kernel<<<BATCH, 256, 0, stream>>>(x_glob, Wq, Wk, wtil);
    logits_kernel<<<dim3(TL / 128, BATCH), 256, 0, stream>>>(x_local, wtil, logits);
    softmax_stats_kernel<<<BATCH * HEADS, 256, 0, stream>>>(logits, msum);
    svec_kernel<<<dim3(8, BATCH), 256, 0, stream>>>(x_local, logits, msum, svec);
    out_kernel<<<BATCH, 256, 0, stream>>>(svec, Wv, Wu, bu, y);
}